// EventSimulator_1675037245805
// MI455X (gfx1250) — compile-verified
//
#include <hip/hip_runtime.h>

// Event-camera simulator: fused log-compression + per-pixel temporal scan.
// [T=64, H=720, W=1280]; memory-bound single-pass stream (~0.95 GB total),
// so the kernel is built around coalesced 128-bit non-temporal traffic.

typedef float v4f __attribute__((ext_vector_type(4)));

#define T_FRAMES 64
#define HPIX     720
#define WPIX     1280
#define NPIX     (HPIX * WPIX)            // 921600, divisible by 4
#define NQUAD    (NPIX / 4)               // 230400

// GAIN = 1 / ln(10000)
#define GAIN_F      0.10857362047581294f
#define DARK_E_F    5.0f
#define DT_F        0.001f
#define REFRACT_F   0.002f

__global__ __launch_bounds__(256) void event_sim_kernel(
    const float* __restrict__ stack,       // [T,H,W]
    const float* __restrict__ pos_map,     // [H,W]
    const float* __restrict__ neg_map,     // [H,W]
    const float* __restrict__ noise_map,   // [H,W]
    const float* __restrict__ ref_v0,      // [H,W]
    const float* __restrict__ dark_noise,  // [T,H,W]
    const float* __restrict__ tmp_noise,   // [T,H,W]
    float* __restrict__ out)               // [T,H,W] in {-1,0,+1}
{
    const int gid = blockIdx.x * blockDim.x + threadIdx.x;
    if (gid >= NQUAD) return;
    const long pbase = (long)gid * 4;

    // Per-pixel maps: loaded once, held in registers across the whole scan.
    const v4f pm  = __builtin_nontemporal_load((const v4f*)(pos_map   + pbase));
    const v4f nm  = __builtin_nontemporal_load((const v4f*)(neg_map   + pbase));
    const v4f nz  = __builtin_nontemporal_load((const v4f*)(noise_map + pbase));
    v4f       refv= __builtin_nontemporal_load((const v4f*)(ref_v0    + pbase));

    v4f nlo, nhi, refr;
    #pragma unroll
    for (int j = 0; j < 4; ++j) {
        nlo[j]  = nz[j] * 0.5f;          // leak draw: tmp < noise*0.5  -> pos
        nhi[j]  = 1.0f - nz[j] * 0.5f;   // leak draw: tmp > 1-noise*0.5 -> neg
        refr[j] = 0.0f;                  // refractory_map starts at clock=0
    }

    const float* sp = stack      + pbase;
    const float* dp = dark_noise + pbase;
    const float* tp = tmp_noise  + pbase;
    float*       op = out        + pbase;

    float clock = 0.0f;

    #pragma unroll 4
    for (int t = 0; t < T_FRAMES; ++t) {
        const long off = (long)t * (long)NPIX;
        v4f st = __builtin_nontemporal_load((const v4f*)(sp + off));
        v4f dn = __builtin_nontemporal_load((const v4f*)(dp + off));
        v4f tn = __builtin_nontemporal_load((const v4f*)(tp + off));

        v4f frame;
        #pragma unroll
        for (int j = 0; j < 4; ++j) {
            // v = clip(GAIN * log(stack + DARK_E * randn), 0, 1)
            float cv = GAIN_F * __logf(st[j] + DARK_E_F * dn[j]);
            cv = fminf(fmaxf(cv, 0.0f), 1.0f);

            const bool  ready = (refr[j] <= clock);
            const float diff  = cv - refv[j];
            bool pos = (diff > pm[j]) && ready;
            bool neg = (diff < nm[j]) && ready;

            // pixels that already fired are excluded from leak draws
            const float tmp = (pos || neg) ? 0.5f : tn[j];
            pos = pos || (tmp < nlo[j]);
            neg = neg || (tmp > nhi[j]);

            frame[j] = (pos ? 1.0f : 0.0f) - (neg ? 1.0f : 0.0f);

            if (pos || neg) {
                refv[j] = cv;
                refr[j] = clock + REFRACT_F;
            }
        }
        __builtin_nontemporal_store(frame, (v4f*)(op + off));
        clock += DT_F;   // replicate float32 accumulation of the jax scan
    }
}

extern "C" void kernel_launch(void* const* d_in, const int* in_sizes, int n_in,
                              void* d_out, int out_size, void* d_ws, size_t ws_size,
                              hipStream_t stream) {
    // setup_inputs() order:
    //   0: stack [T,H,W]  1: pos_map [H,W]  2: neg_map [H,W]  3: noise_map [H,W]
    //   4: ref_v0 [H,W]   5: dark_noise [T,H,W]   6: tmp_noise [T,H,W]
    const float* stack      = (const float*)d_in[0];
    const float* pos_map    = (const float*)d_in[1];
    const float* neg_map    = (const float*)d_in[2];
    const float* noise_map  = (const float*)d_in[3];
    const float* ref_v0     = (const float*)d_in[4];
    const float* dark_noise = (const float*)d_in[5];
    const float* tmp_noise  = (const float*)d_in[6];
    float* out = (float*)d_out;

    const int threads = 256;
    const int blocks  = (NQUAD + threads - 1) / threads;  // 900
    event_sim_kernel<<<blocks, threads, 0, stream>>>(
        stack, pos_map, neg_map, noise_map, ref_v0, dark_noise, tmp_noise, out);
}